// MultiScaleRoIAlign_8753143349405
// MI455X (gfx1250) — compile-verified
//
#include <hip/hip_runtime.h>
#include <stdint.h>

#define NCH 256
#define NROI 512

typedef unsigned int v4u __attribute__((ext_vector_type(4)));
typedef int v8i __attribute__((ext_vector_type(8)));
typedef int v4i __attribute__((ext_vector_type(4)));

__device__ __forceinline__ int imin(int a, int b) { return a < b ? a : b; }

// torchvision roi_align (aligned=false) coordinate prep
__device__ __forceinline__ void prep_coord(float c, int size, int& lo, int& hi,
                                           float& frac, float& valid) {
    valid = (c > -1.0f && c < (float)size) ? 1.0f : 0.0f;
    c = fmaxf(c, 0.0f);
    int l = imin((int)floorf(c), size - 1);
    lo = l;
    hi = imin(l + 1, size - 1);
    frac = (l >= size - 1) ? 0.0f : (c - (float)l);
}

// Per-ROI setup: level assignment + bin geometry (uniform across block).
__device__ __forceinline__ void roi_ctx(const float* __restrict__ b0,
                                        const float* __restrict__ b1, int r,
                                        int& b, int& lvl, int& W, int& P,
                                        float& x1s, float& y1s,
                                        float& bin_w, float& bin_h) {
    const float* box = (r < 256) ? (b0 + 4 * r) : (b1 + 4 * (r - 256));
    b = (r < 256) ? 0 : 1;
    float x1 = box[0], y1 = box[1], x2 = box[2], y2 = box[3];
    float s = sqrtf((x2 - x1) * (y2 - y1));
    float tgt = floorf(4.0f + log2f(s / 224.0f) + 1e-6f);
    tgt = fminf(fmaxf(tgt, 2.0f), 5.0f);
    lvl = (int)tgt - 2;
    W = 200 >> lvl;           // 200,100,50,25 (square)
    P = W * W;
    float scale = 0.25f / (float)(1 << lvl);
    x1s = x1 * scale; y1s = y1 * scale;
    float x2s = x2 * scale, y2s = y2 * scale;
    bin_w = fmaxf(x2s - x1s, 1.0f) * (1.0f / 7.0f);
    bin_h = fmaxf(y2s - y1s, 1.0f) * (1.0f / 7.0f);
}

// Build the 196-subsample offset/weight tables in LDS. elemMul = channel pitch
// of one spatial element (NCH for NHWC layout, 1 for NCHW layout).
__device__ __forceinline__ void build_tables(int W, float x1s, float y1s,
                                             float bin_w, float bin_h, int elemMul,
                                             int4* s_o, float4* s_w) {
    for (int s = threadIdx.x; s < 196; s += blockDim.x) {
        int sy = s / 14, sx = s % 14;
        float y = y1s + ((float)sy + 0.5f) * 0.5f * bin_h;
        float x = x1s + ((float)sx + 0.5f) * 0.5f * bin_w;
        int yl, yh, xl, xh; float fy, fx, vy, vx;
        prep_coord(y, W, yl, yh, fy, vy);
        prep_coord(x, W, xl, xh, fx, vx);
        float w = vy * vx * 0.25f;                    // fold 2x2-subsample average
        float4 wt;
        wt.x = (1.0f - fy) * (1.0f - fx) * w;
        wt.y = (1.0f - fy) * fx * w;
        wt.z = fy * (1.0f - fx) * w;
        wt.w = fy * fx * w;
        int4 of;
        of.x = (yl * W + xl) * elemMul;
        of.y = (yl * W + xh) * elemMul;
        of.z = (yh * W + xl) * elemMul;
        of.w = (yh * W + xh) * elemMul;
        s_o[s] = of;
        s_w[s] = wt;
    }
}

// ---------------------------------------------------------------------------
// Kernel 1: TDM-based NCHW -> NHWC transpose.  One tensor_load_to_lds per
// 32(c) x 128(p) tile: row pitch tensor_dim0_stride = P elements, LDS pad of
// 5 dwords per 128-dword row => 133-dword LDS row pitch (odd => conflict-free
// transposed reads).  Wave 0 issues the TDM op and waits TENSORcnt, then the
// whole workgroup writes coalesced NHWC stores.
// ---------------------------------------------------------------------------
__global__ void tdm_transpose_nchw_to_nhwc(const float* __restrict__ in,
                                           float* __restrict__ out, int P) {
    __shared__ float tile[32 * 133];
    int p0 = blockIdx.x * 128;
    if (p0 > P - 128) p0 = P - 128;          // overlap-clamped edge tile
    int c0 = blockIdx.y * 32;
    int b  = blockIdx.z;

    const float* src = in + ((size_t)(b * NCH + c0)) * (size_t)P + p0;

    if (threadIdx.x < 32) {                  // wave 0 only: issue the TDM op
        uint64_t ga = (uint64_t)(uintptr_t)src;
        unsigned lds = (unsigned)(uintptr_t)&tile[0];   // generic LDS addr low 32 = offset
        // D# group 0: count=1 | lds_addr | global_addr[56:0] | type=2
        v4u g0;
        g0.x = 1u;
        g0.y = lds;
        g0.z = (unsigned)(ga & 0xFFFFFFFFu);
        g0.w = (unsigned)((ga >> 32) & 0x01FFFFFFu) | 0x80000000u;
        // D# group 1: data_size=4B(2), pad_enable, pad_interval=128dw(6),
        // pad_amount=5dw(4); tensor dims huge (tiles pre-clamped in range);
        // tile_dim0=128 (p, contiguous), tile_dim1=32 (c), dim0_stride=P.
        const unsigned TD0 = 0x40000000u, TD1 = 0x40000000u;
        v8i g1;
        g1[0] = (int)((2u << 16) | (1u << 20) | (6u << 22) | (4u << 25));
        g1[1] = (int)((TD0 & 0xFFFFu) << 16);
        g1[2] = (int)(((TD0 >> 16) & 0xFFFFu) | ((TD1 & 0xFFFFu) << 16));
        g1[3] = (int)(((TD1 >> 16) & 0xFFFFu) | (128u << 16));
        g1[4] = 32;                      // tile_dim1=32, tile_dim2=0
        g1[5] = P;                       // tensor_dim0_stride[31:0]
        g1[6] = 0;
        g1[7] = 0;
        v4i g2 = {0, 0, 0, 0};
        v4i g3 = {0, 0, 0, 0};
        v8i g4 = {0, 0, 0, 0, 0, 0, 0, 0};
        __builtin_amdgcn_tensor_load_to_lds(g0, g1, g2, g3, g4, 0);
        __builtin_amdgcn_s_wait_tensorcnt(0);
    }
    __syncthreads();

    int cl = threadIdx.x & 31;           // local channel
    int ty = threadIdx.x >> 5;
#pragma unroll
    for (int i = 0; i < 16; ++i) {
        int pl = ty + 8 * i;             // local p 0..127
        out[((size_t)b * (size_t)P + (size_t)(p0 + pl)) * NCH + (c0 + cl)] =
            tile[cl * 133 + pl];
    }
}

// ---------------------------------------------------------------------------
// Kernel 2: gather from NHWC copy.  Block = 1 ROI, 64 threads; each thread
// owns 4 consecutive channels => float4 (global_load_b128) fully coalesced
// across the wave.  Offset/weight tables broadcast from LDS.
// ---------------------------------------------------------------------------
__global__ void roi_gather_nhwc(const float* __restrict__ nhwc,
                                const float* __restrict__ b0,
                                const float* __restrict__ b1,
                                float* __restrict__ out) {
    __shared__ int4 s_o[196];
    __shared__ float4 s_w[196];

    int r = blockIdx.x;
    int b, lvl, W, P; float x1s, y1s, bin_w, bin_h;
    roi_ctx(b0, b1, r, b, lvl, W, P, x1s, y1s, bin_w, bin_h);
    build_tables(W, x1s, y1s, bin_w, bin_h, NCH, s_o, s_w);
    __syncthreads();

    size_t lvl_off;
    switch (lvl) {
        case 0:  lvl_off = 0ull;          break;
        case 1:  lvl_off = 20480000ull;   break;
        case 2:  lvl_off = 25600000ull;   break;
        default: lvl_off = 26880000ull;   break;
    }
    int c = 4 * threadIdx.x;
    const float* base = nhwc + lvl_off + (size_t)b * (size_t)P * NCH + c;
    size_t obase = ((size_t)r * NCH + c) * 49;

    for (int by = 0; by < 7; ++by) {
        for (int bx = 0; bx < 7; ++bx) {
            float ax = 0.f, ay = 0.f, az = 0.f, aw = 0.f;
#pragma unroll
            for (int jy = 0; jy < 2; ++jy) {
#pragma unroll
                for (int jx = 0; jx < 2; ++jx) {
                    int s = (2 * by + jy) * 14 + (2 * bx + jx);
                    int4 o = s_o[s];
                    float4 w = s_w[s];
                    float4 v;
                    v = *(const float4*)(base + o.x);
                    ax += w.x * v.x; ay += w.x * v.y; az += w.x * v.z; aw += w.x * v.w;
                    v = *(const float4*)(base + o.y);
                    ax += w.y * v.x; ay += w.y * v.y; az += w.y * v.z; aw += w.y * v.w;
                    v = *(const float4*)(base + o.z);
                    ax += w.z * v.x; ay += w.z * v.y; az += w.z * v.z; aw += w.z * v.w;
                    v = *(const float4*)(base + o.w);
                    ax += w.w * v.x; ay += w.w * v.y; az += w.w * v.z; aw += w.w * v.w;
                }
            }
            int bin = by * 7 + bx;
            out[obase + bin]           = ax;
            out[obase + 49 + bin]      = ay;
            out[obase + 2 * 49 + bin]  = az;
            out[obase + 3 * 49 + bin]  = aw;
        }
    }
}

// ---------------------------------------------------------------------------
// Kernel 3: fallback direct NCHW gather (used if workspace too small for the
// NHWC copy).  Block = 1 ROI, 256 threads = 1 channel each.
// ---------------------------------------------------------------------------
__global__ void roi_gather_nchw(const float* __restrict__ f0,
                                const float* __restrict__ f1,
                                const float* __restrict__ f2,
                                const float* __restrict__ f3,
                                const float* __restrict__ b0,
                                const float* __restrict__ b1,
                                float* __restrict__ out) {
    __shared__ int4 s_o[196];
    __shared__ float4 s_w[196];

    int r = blockIdx.x;
    int b, lvl, W, P; float x1s, y1s, bin_w, bin_h;
    roi_ctx(b0, b1, r, b, lvl, W, P, x1s, y1s, bin_w, bin_h);
    build_tables(W, x1s, y1s, bin_w, bin_h, 1, s_o, s_w);
    __syncthreads();

    const float* f = (lvl == 0) ? f0 : (lvl == 1) ? f1 : (lvl == 2) ? f2 : f3;
    int c = threadIdx.x;
    const float* base = f + ((size_t)(b * NCH + c)) * (size_t)P;
    size_t obase = ((size_t)r * NCH + c) * 49;

    for (int by = 0; by < 7; ++by) {
        for (int bx = 0; bx < 7; ++bx) {
            float acc = 0.f;
#pragma unroll
            for (int jy = 0; jy < 2; ++jy) {
#pragma unroll
                for (int jx = 0; jx < 2; ++jx) {
                    int s = (2 * by + jy) * 14 + (2 * bx + jx);
                    int4 o = s_o[s];
                    float4 w = s_w[s];
                    acc += w.x * base[o.x] + w.y * base[o.y] +
                           w.z * base[o.z] + w.w * base[o.w];
                }
            }
            out[obase + by * 7 + bx] = acc;
        }
    }
}

extern "C" void kernel_launch(void* const* d_in, const int* in_sizes, int n_in,
                              void* d_out, int out_size, void* d_ws, size_t ws_size,
                              hipStream_t stream) {
    const float* f[4] = {(const float*)d_in[0], (const float*)d_in[1],
                         (const float*)d_in[2], (const float*)d_in[3]};
    const float* b0 = (const float*)d_in[4];
    const float* b1 = (const float*)d_in[5];
    float* out = (float*)d_out;

    static const int    Ps[4]   = {40000, 10000, 2500, 625};
    static const size_t offs[4] = {0ull, 20480000ull, 25600000ull, 26880000ull};
    const size_t need = 27200000ull * sizeof(float);   // 108.8 MB NHWC copy

    if (ws_size >= need) {
        float* nhwc = (float*)d_ws;
        for (int l = 0; l < 4; ++l) {
            int P = Ps[l];
            dim3 grid((P + 127) / 128, NCH / 32, 2);
            hipLaunchKernelGGL(tdm_transpose_nchw_to_nhwc, grid, dim3(256), 0,
                               stream, f[l], nhwc + offs[l], P);
        }
        hipLaunchKernelGGL(roi_gather_nhwc, dim3(NROI), dim3(64), 0, stream,
                           nhwc, b0, b1, out);
    } else {
        hipLaunchKernelGGL(roi_gather_nchw, dim3(NROI), dim3(256), 0, stream,
                           f[0], f[1], f[2], f[3], b0, b1, out);
    }
}